// HeadController_70334384439966
// MI455X (gfx1250) — compile-verified
//
#include <hip/hip_runtime.h>

// ---- problem constants (match reference) ----
#define B_    2
#define CIN_  48
#define CMID_ 8
#define G_    16
#define CPG_  3
#define HID_  256
#define P_    153
#define S_    884736            // 96*96*96
#define R_    (CPG_ * S_)       // elements per (b,g) group = 2654208
#define CCLS_ 16
#define EPS_  1e-5f

// ---- workspace layout (float offsets) ----
#define WS_STATS  0             // 2 * B * G   = 64   (sum, sumsq) pairs
#define WS_SCALE  64            // B * CIN     = 96
#define WS_SHIFT  160           // B * CIN     = 96
#define WS_PARAMS 256           // B * P       = 306

typedef float v2f __attribute__((ext_vector_type(2)));
typedef float v8f __attribute__((ext_vector_type(8)));

// ---------------------------------------------------------------------------
// Kernel 1: controller GEMV  params[b,p] = ctrl_b[p] + sum_h x[b,h]*ctrl_w[p,h]
// ---------------------------------------------------------------------------
__global__ void k_ctrl(const float* __restrict__ x, const float* __restrict__ w,
                       const float* __restrict__ bias, float* __restrict__ wsF) {
    __shared__ float sx[HID_];
    const int b = blockIdx.x;
    const int t = threadIdx.x;                 // 256 threads
    sx[t] = x[b * HID_ + t];
    __syncthreads();
    if (t < P_) {
        const float* wr = w + (size_t)t * HID_;
        float acc = bias[t];
        #pragma unroll 8
        for (int h = 0; h < HID_; ++h) acc = fmaf(sx[h], wr[h], acc);
        wsF[WS_PARAMS + b * P_ + t] = acc;
    }
}

// ---------------------------------------------------------------------------
// Kernel 2: GroupNorm partial stats. Region (b,g) is R_ contiguous floats.
// float4 (B128) streaming loads: 6912 float4/block = 256 thr * 27 exactly.
// ---------------------------------------------------------------------------
#define STAT_CHUNKS 96
#define STAT_TPB    256
__global__ void k_stats(const float* __restrict__ feat, float* __restrict__ wsF) {
    const int region = blockIdx.x / STAT_CHUNKS;   // b*G + g
    const int chunk  = blockIdx.x % STAT_CHUNKS;
    const float4* base = (const float4*)(feat + (size_t)region * R_);
    const int per4 = R_ / (STAT_CHUNKS * 4);       // 6912 float4 per block
    const int end  = (chunk + 1) * per4;
    float s = 0.f, q = 0.f;
    for (int i = chunk * per4 + threadIdx.x; i < end; i += STAT_TPB) {
        float4 v = base[i];
        s += v.x + v.y + v.z + v.w;
        q = fmaf(v.x, v.x, q); q = fmaf(v.y, v.y, q);
        q = fmaf(v.z, v.z, q); q = fmaf(v.w, v.w, q);
    }
    // wave32 reduction
    for (int off = 16; off; off >>= 1) {
        s += __shfl_down(s, off);
        q += __shfl_down(q, off);
    }
    __shared__ float ls[STAT_TPB / 32], lq[STAT_TPB / 32];
    const int lane = threadIdx.x & 31, wv = threadIdx.x >> 5;
    if (lane == 0) { ls[wv] = s; lq[wv] = q; }
    __syncthreads();
    if (threadIdx.x == 0) {
        float S2 = 0.f, Q2 = 0.f;
        #pragma unroll
        for (int i = 0; i < STAT_TPB / 32; ++i) { S2 += ls[i]; Q2 += lq[i]; }
        atomicAdd(&wsF[WS_STATS + 2 * region + 0], S2);
        atomicAdd(&wsF[WS_STATS + 2 * region + 1], Q2);
    }
}

// ---------------------------------------------------------------------------
// Kernel 3: finalize stats -> folded per-channel scale/shift
// ---------------------------------------------------------------------------
__global__ void k_final(const float* __restrict__ gamma, const float* __restrict__ beta,
                        float* __restrict__ wsF) {
    __shared__ float s_mean[B_ * G_], s_rstd[B_ * G_];
    const int t = threadIdx.x;                 // 128 threads
    if (t < B_ * G_) {
        const float n = (float)R_;
        float m = wsF[WS_STATS + 2 * t + 0] / n;
        float v = wsF[WS_STATS + 2 * t + 1] / n - m * m;
        s_mean[t] = m;
        s_rstd[t] = rsqrtf(v + EPS_);
    }
    __syncthreads();
    if (t < B_ * CIN_) {
        const int b = t / CIN_, c = t % CIN_, g = c / CPG_;
        const int idx = b * G_ + g;
        const float sc = gamma[c] * s_rstd[idx];
        wsF[WS_SCALE + t] = sc;
        wsF[WS_SHIFT + t] = fmaf(-s_mean[idx], sc, beta[c]);
    }
}

// ---------------------------------------------------------------------------
// Kernel 4: fused normalize + ReLU + 8x48 projection (WMMA f32 16x16x4)
//           + per-batch 8->8->8->1 MLP + broadcast store to 16 channels.
// One wave handles 32 voxels per tile iteration (two 16-voxel WMMA halves).
// Load phase stages all 48 values (long load clauses, 48 loads in flight per
// wave); __launch_bounds__(.,2) relaxes the VGPR cap so staging + A fragments
// + accumulators live entirely in registers (no scratch spills).
// ---------------------------------------------------------------------------
#define WAVES_PB 8
#define MAIN_TPB (WAVES_PB * 32)
#define TILES_   (S_ / 32)                     // 27648 per batch

__global__ void __launch_bounds__(MAIN_TPB, 2)
k_main(const float* __restrict__ feat, const float* __restrict__ wsF,
       const float* __restrict__ pre_w, const float* __restrict__ pre_b,
       float* __restrict__ out) {
    __shared__ float s_sc[CIN_], s_sh[CIN_], s_pb[CMID_], s_pp[P_];
    const int b   = blockIdx.y;
    const int tid = threadIdx.x;
    if (tid < CIN_)  { s_sc[tid] = wsF[WS_SCALE + b * CIN_ + tid];
                       s_sh[tid] = wsF[WS_SHIFT + b * CIN_ + tid]; }
    if (tid < CMID_)   s_pb[tid] = pre_b[tid];
    if (tid < P_)      s_pp[tid] = wsF[WS_PARAMS + b * P_ + tid];
    __syncthreads();

    const int lane = tid & 31;
    const int n    = lane & 15;                // N index / A-matrix row
    const int koff = (lane < 16) ? 0 : 2;      // K sub-offset for this lane half

    // A fragments: pre_w (8x48) zero-padded to 16x48, 12 chunks of K=4.
    // 32-bit A 16x4 layout: VGPR0 = A[m][k4 + (lane<16?0:2)], VGPR1 = +1.
    v2f A[12];
    #pragma unroll
    for (int k = 0; k < 12; ++k) {
        float a0 = 0.f, a1 = 0.f;
        if (n < CMID_) {
            a0 = pre_w[n * CIN_ + 4 * k + koff + 0];
            a1 = pre_w[n * CIN_ + 4 * k + koff + 1];
        }
        v2f a; a.x = a0; a.y = a1;
        A[k] = a;
    }

    // lane-specific base: batch + K-sub-offset channel + N index
    const float* fb  = feat + (size_t)b * CIN_ * S_ + (size_t)koff * S_ + n;
    float*       ob  = out  + (size_t)b * CCLS_ * S_;
    const int waveSlot   = blockIdx.x * WAVES_PB + (tid >> 5);
    const int waveStride = gridDim.x * WAVES_PB;

    for (int tile = waveSlot; tile < TILES_; tile += waveStride) {
        const int v0 = tile * 32;

        // ---- load phase: all 48 values staged, long clauses, one drain ----
        float x00[12], x01[12], x10[12], x11[12];
        #pragma unroll
        for (int k = 0; k < 12; ++k) {
            const float* p0 = fb + (size_t)(4 * k) * S_ + v0;  // channel 4k+koff
            const float* p1 = p0 + S_;                         // channel 4k+koff+1
            x00[k] = __builtin_nontemporal_load(p0);           // last-use stream
            x01[k] = __builtin_nontemporal_load(p1);
            x10[k] = __builtin_nontemporal_load(p0 + 16);
            x11[k] = __builtin_nontemporal_load(p1 + 16);
        }

        // ---- compute phase: normalize+ReLU into B frags, 24 WMMAs ----
        v8f acc0 = {};                         // voxels v0 .. v0+15
        v8f acc1 = {};                         // voxels v0+16 .. v0+31
        #pragma unroll
        for (int k = 0; k < 12; ++k) {
            const int c0 = 4 * k + koff, c1 = c0 + 1;
            const float sc0 = s_sc[c0], sh0 = s_sh[c0];
            const float sc1 = s_sc[c1], sh1 = s_sh[c1];
            v2f B0, B1;                        // B 4x16 layout mirrors A striping
            B0.x = fmaxf(fmaf(x00[k], sc0, sh0), 0.f);
            B0.y = fmaxf(fmaf(x01[k], sc1, sh1), 0.f);
            B1.x = fmaxf(fmaf(x10[k], sc0, sh0), 0.f);
            B1.y = fmaxf(fmaf(x11[k], sc1, sh1), 0.f);
            acc0 = __builtin_amdgcn_wmma_f32_16x16x4_f32(false, A[k], false, B0,
                                                         (short)0, acc0, false, false);
            acc1 = __builtin_amdgcn_wmma_f32_16x16x4_f32(false, A[k], false, B1,
                                                         (short)0, acc1, false, false);
        }

        // D layout: lane n (0..15), VGPR m  ->  (M=m, N=n). Redistribute so
        // lane L owns y8 of voxel v0+L (tile half = L/16, src lane = L%16).
        float y[CMID_];
        #pragma unroll
        for (int i = 0; i < CMID_; ++i) {
            const float t0 = __shfl(acc0[i], n, 32);
            const float t1 = __shfl(acc1[i], n, 32);
            y[i] = ((lane < 16) ? t0 : t1) + s_pb[i];
        }
        // dynamic per-batch MLP: relu(W1 y + b1) -> relu(W2 . + b2) -> w3 . + b3
        float h1[CMID_], h2[CMID_];
        #pragma unroll
        for (int o = 0; o < CMID_; ++o) {
            float a = s_pp[136 + o];
            #pragma unroll
            for (int i = 0; i < CMID_; ++i) a = fmaf(s_pp[o * 8 + i], y[i], a);
            h1[o] = fmaxf(a, 0.f);
        }
        #pragma unroll
        for (int o = 0; o < CMID_; ++o) {
            float a = s_pp[144 + o];
            #pragma unroll
            for (int i = 0; i < CMID_; ++i) a = fmaf(s_pp[64 + o * 8 + i], h1[i], a);
            h2[o] = fmaxf(a, 0.f);
        }
        float o3 = s_pp[152];
        #pragma unroll
        for (int i = 0; i < CMID_; ++i) o3 = fmaf(s_pp[128 + i], h2[i], o3);

        // broadcast scalar to all 16 class channels, streaming stores
        float* op = ob + v0 + lane;
        #pragma unroll
        for (int cc = 0; cc < CCLS_; ++cc)
            __builtin_nontemporal_store(o3, op + (size_t)cc * S_);
    }
}

// ---------------------------------------------------------------------------
extern "C" void kernel_launch(void* const* d_in, const int* in_sizes, int n_in,
                              void* d_out, int out_size, void* d_ws, size_t ws_size,
                              hipStream_t stream) {
    const float* x     = (const float*)d_in[0];
    const float* feat  = (const float*)d_in[1];
    const float* cw    = (const float*)d_in[2];
    const float* cb    = (const float*)d_in[3];
    const float* gamma = (const float*)d_in[4];
    const float* beta  = (const float*)d_in[5];
    const float* pw    = (const float*)d_in[6];
    const float* pb    = (const float*)d_in[7];
    float* out = (float*)d_out;
    float* wsF = (float*)d_ws;

    (void)in_sizes; (void)n_in; (void)out_size; (void)ws_size;

    // zero only the atomic stats accumulators (ws is poisoned by harness)
    hipMemsetAsync(wsF + WS_STATS, 0, 2 * B_ * G_ * sizeof(float), stream);

    k_ctrl <<<B_, 256, 0, stream>>>(x, cw, cb, wsF);
    k_stats<<<B_ * G_ * STAT_CHUNKS, STAT_TPB, 0, stream>>>(feat, wsF);
    k_final<<<1, 128, 0, stream>>>(gamma, beta, wsF);

    dim3 grid(432, B_);                         // 432*8 waves * 8 tiles = 27648/batch
    k_main<<<grid, MAIN_TPB, 0, stream>>>(feat, wsF, pw, pb, out);
}